// MutualInfoEstimator_27169963115009
// MI455X (gfx1250) — compile-verified
//
#include <hip/hip_runtime.h>
#include <hip/hip_bf16.h>

#define DIM  256
#define HID  512
#define NPTS 1024

typedef __attribute__((ext_vector_type(2))) float v2f;
typedef __attribute__((ext_vector_type(8))) float v8f;

// ---------------------------------------------------------------------------
// One wave32 computes a 16x64 strip (4 adjacent 16x16 tiles) of
//   Out = act(A @ W + bias)          (CLUB == false)
// or fuses the CLUB closed-form reduction over its strip of mu = A@W + bias
// directly in the epilogue (CLUB == true; nothing is stored, one atomicAdd
// per wave).
//
// A is logically [NPTS, K]: columns [0,K0) from A0 (row stride K0), columns
// [K0,K) from A1 (row stride K-K0). Non-concat: pass K0 == K. W is [K, Nout]
// row-major. For CLUB, Nout == DIM and y/cs/cs2 are [NPTS,DIM]/[DIM]/[DIM].
// ---------------------------------------------------------------------------
template <bool RELU, bool CLUB>
__global__ __launch_bounds__(256) void wmma_gemm_fused(
    const float* __restrict__ A0, const float* __restrict__ A1, int K0,
    const float* __restrict__ W,  const float* __restrict__ bias,
    float* __restrict__ Out, int K, int Nout,
    const float* __restrict__ y,  const float* __restrict__ cs,
    const float* __restrict__ cs2, float* __restrict__ accp)
{
    const int wave = threadIdx.x >> 5;
    const int lane = threadIdx.x & 31;
    const int half = lane >> 4;          // 16-lane half of the wave
    const int l16  = lane & 15;
    const int stripsN = Nout >> 6;       // 64-wide strips
    const int t   = blockIdx.x * 8 + wave;
    const int m0  = (t / stripsN) << 4;
    const int n0  = (t % stripsN) << 6;
    const int mrow = m0 + l16;

    v8f acc[4] = {{}, {}, {}, {}};

#if defined(__HIP_DEVICE_COMPILE__)
    // V_WMMA_F32_16X16X4_F32 layouts:
    //   A 16x4 : lane l16 = M, K = 2*half + vgpr  -> one contiguous b64 load
    //   B 4x16 : K = 2*half + vgpr, lane l16 = N
    //   C/D    : vgpr v -> M = v + 8*half, lane l16 = N
    {
        const float* __restrict__ wp = W + (size_t)(2 * half) * Nout + n0 + l16;
        const float* __restrict__ ap = A0 + (size_t)mrow * K0 + 2 * half;
        for (int k = 0; k < K0; k += 4) {
            v2f av;
            av[0] = ap[0];
            av[1] = ap[1];
#pragma unroll
            for (int j = 0; j < 4; ++j) {
                v2f bv;
                bv[0] = wp[16 * j];
                bv[1] = wp[(size_t)Nout + 16 * j];
                acc[j] = __builtin_amdgcn_wmma_f32_16x16x4_f32(
                    false, av, false, bv, (short)0, acc[j], false, false);
            }
            ap += 4;
            wp += (size_t)4 * Nout;
        }
        if (K > K0) {   // concat second source; wp keeps advancing through W
            const int K1 = K - K0;
            ap = A1 + (size_t)mrow * K1 + 2 * half;
            for (int k = K0; k < K; k += 4) {
                v2f av;
                av[0] = ap[0];
                av[1] = ap[1];
#pragma unroll
                for (int j = 0; j < 4; ++j) {
                    v2f bv;
                    bv[0] = wp[16 * j];
                    bv[1] = wp[(size_t)Nout + 16 * j];
                    acc[j] = __builtin_amdgcn_wmma_f32_16x16x4_f32(
                        false, av, false, bv, (short)0, acc[j], false, false);
                }
                ap += 4;
                wp += (size_t)4 * Nout;
            }
        }
    }
#endif

    if constexpr (!CLUB) {
#pragma unroll
        for (int j = 0; j < 4; ++j) {
            const int n = n0 + 16 * j + l16;
            const float bb = bias[n];
#pragma unroll
            for (int v = 0; v < 8; ++v) {
                const int m = m0 + v + 8 * half;
                float val = acc[j][v] + bb;
                if (RELU) val = fmaxf(val, 0.0f);
                Out[(size_t)m * Nout + n] = val;
            }
        }
    } else {
        // CLUB closed form per element (i=m, d=n):
        //   -0.5*(mu-y)^2 + (Sy2[d] - 2*mu*Sy[d] + N*mu^2)/(2N)
        float local = 0.0f;
#pragma unroll
        for (int j = 0; j < 4; ++j) {
            const int n = n0 + 16 * j + l16;
            const float bb = bias[n];
            const float c1 = cs[n];
            const float c2 = cs2[n];
#pragma unroll
            for (int v = 0; v < 8; ++v) {
                const int m = m0 + v + 8 * half;
                const float mu = acc[j][v] + bb;
                const float dp = mu - y[(size_t)m * Nout + n];
                local += -0.5f * dp * dp
                       + 0.5f * (c2 - 2.0f * mu * c1 + (float)NPTS * mu * mu)
                              * (1.0f / (float)NPTS);
            }
        }
        // wave32 reduction, then one atomic per wave
        for (int o = 16; o > 0; o >>= 1) local += __shfl_xor(local, o, 32);
        if (lane == 0) atomicAdd(accp, local);
    }
}

// ---------------------------------------------------------------------------
// Column sums / sums-of-squares of y [NPTS, DIM], parallel over row blocks.
// cs/cs2 must be zeroed first; 32 blocks each reduce 32 rows then atomicAdd.
// ---------------------------------------------------------------------------
#define CS_ROWS 32
__global__ __launch_bounds__(256) void colstats_kernel(
    const float* __restrict__ y, float* __restrict__ cs, float* __restrict__ cs2)
{
    const int d  = threadIdx.x;           // 256 == DIM
    const int j0 = blockIdx.x * CS_ROWS;
    float s = 0.0f, s2 = 0.0f;
    for (int j = j0; j < j0 + CS_ROWS; ++j) {
        const float v = y[(size_t)j * DIM + d];
        s  += v;
        s2 += v * v;
    }
    atomicAdd(&cs[d], s);
    atomicAdd(&cs2[d], s2);
}

__global__ __launch_bounds__(256) void zero_ws_kernel(float* __restrict__ ws)
{
    for (int i = threadIdx.x; i < 1040; i += 256) ws[i] = 0.0f;
}

__global__ void finalize_kernel(const float* __restrict__ acc, float* __restrict__ out)
{
    out[0] = (acc[0] + acc[1]) * (1.0f / (float)NPTS);
}

// ---------------------------------------------------------------------------
extern "C" void kernel_launch(void* const* d_in, const int* in_sizes, int n_in,
                              void* d_out, int out_size, void* d_ws, size_t ws_size,
                              hipStream_t stream)
{
    (void)in_sizes; (void)n_in; (void)out_size; (void)ws_size;

    const float* hist = (const float*)d_in[0];
    const float* path = (const float*)d_in[1];
    const float* glob = (const float*)d_in[2];
    const float* W1   = (const float*)d_in[3];
    const float* b1   = (const float*)d_in[4];
    const float* W2   = (const float*)d_in[5];
    const float* b2   = (const float*)d_in[6];
    const float* W1g  = (const float*)d_in[7];
    const float* b1g  = (const float*)d_in[8];
    const float* W2g  = (const float*)d_in[9];
    const float* b2g  = (const float*)d_in[10];
    float* out = (float*)d_out;

    // Workspace (floats): [acc x2 | pad..16 | cs_p 256 | cs2_p 256 |
    //                      cs_g 256 | cs2_g 256 | H 1024*512]  ~= 2.1 MB
    float* ws    = (float*)d_ws;
    float* acc   = ws;
    float* cs_p  = ws + 16;
    float* cs2_p = ws + 16 + 256;
    float* cs_g  = ws + 16 + 512;
    float* cs2_g = ws + 16 + 768;
    float* Hbuf  = ws + 1040;                 // [NPTS, HID]

    const int l1_blocks = (NPTS / 16) * (HID / 64) / 8;   // 64
    const int l2_blocks = (NPTS / 16) * (DIM / 64) / 8;   // 32
    const int cs_blocks = NPTS / CS_ROWS;                 // 32

    zero_ws_kernel<<<1, 256, 0, stream>>>(ws);
    colstats_kernel<<<cs_blocks, 256, 0, stream>>>(path, cs_p, cs2_p);
    colstats_kernel<<<cs_blocks, 256, 0, stream>>>(glob, cs_g, cs2_g);

    // --- branch 1: mu = MLP(histology); club(mu, pathways) fused ---
    wmma_gemm_fused<true, false><<<l1_blocks, 256, 0, stream>>>(
        hist, hist, DIM, W1, b1, Hbuf, DIM, HID,
        nullptr, nullptr, nullptr, nullptr);
    wmma_gemm_fused<false, true><<<l2_blocks, 256, 0, stream>>>(
        Hbuf, Hbuf, HID, W2, b2, nullptr, HID, DIM,
        path, cs_p, cs2_p, acc + 0);

    // --- branch 2: mug = MLP([hist|path]); club(mug, global_embed) fused ---
    wmma_gemm_fused<true, false><<<l1_blocks, 256, 0, stream>>>(
        hist, path, DIM, W1g, b1g, Hbuf, 2 * DIM, HID,
        nullptr, nullptr, nullptr, nullptr);
    wmma_gemm_fused<false, true><<<l2_blocks, 256, 0, stream>>>(
        Hbuf, Hbuf, HID, W2g, b2g, nullptr, HID, DIM,
        glob, cs_g, cs2_g, acc + 1);

    finalize_kernel<<<1, 1, 0, stream>>>(acc, out);
}